// GCN_30717606101013
// MI455X (gfx1250) — compile-verified
//
#include <hip/hip_runtime.h>
#include <hip/hip_bf16.h>

typedef float v2f __attribute__((ext_vector_type(2)));
typedef float v8f __attribute__((ext_vector_type(8)));

#define IN_C 128
#define HID_C 64

// ---------------- degree / norm ----------------
__global__ void k_deg_init(float* deg, int n) {
  int i = blockIdx.x * blockDim.x + threadIdx.x;
  if (i < n) deg[i] = 1.0f;  // self-loop
}

__global__ void k_deg_edges(const long long* __restrict__ ei, float* deg, int E) {
  int e = blockIdx.x * blockDim.x + threadIdx.x;
  if (e < E) atomicAdd(&deg[(int)ei[(size_t)E + e]], 1.0f);  // dst row
}

__global__ void k_dinv(float* deg_dinv, int n) {
  int i = blockIdx.x * blockDim.x + threadIdx.x;
  if (i < n) deg_dinv[i] = rsqrtf(deg_dinv[i]);  // deg >= 1 always
}

// ---------------- h = x @ W1 via V_WMMA_F32_16X16X4_F32 ----------------
// One wave -> 16x16 tile of h. Block = 8 waves = 32 rows x 64 cols.
__global__ void k_gemm1_wmma(const float* __restrict__ x,
                             const float* __restrict__ W1,
                             float* __restrict__ h, int n) {
  const int lane = threadIdx.x & 31;
  const int wave = threadIdx.x >> 5;
  const int m0 = (blockIdx.x * 2 + (wave >> 2)) * 16;
  const int n0 = (wave & 3) * 16;
  if (m0 >= n) return;

  const int row   = lane & 15;        // A row / B col within tile
  const int khalf = (lane >> 4) << 1; // 0 or 2: K sub-pair per lane half
  const float* arow = x  + (size_t)(m0 + row) * IN_C + khalf;
  const float* bcol = W1 + (size_t)khalf * HID_C + n0 + row;

  v8f acc = {};
#pragma unroll
  for (int k = 0; k < IN_C; k += 4) {
    v2f a, b;
    float2 av = *(const float2*)(arow + k);   // x[row][k+khalf], x[row][k+khalf+1]
    a.x = av.x; a.y = av.y;
    b.x = bcol[(size_t)k * HID_C];            // W1[k+khalf][col]
    b.y = bcol[(size_t)(k + 1) * HID_C];      // W1[k+khalf+1][col]
    acc = __builtin_amdgcn_wmma_f32_16x16x4_f32(
        /*neg_a=*/false, a, /*neg_b=*/false, b,
        /*c_mod=*/(short)0, acc, /*reuse_a=*/false, /*reuse_b=*/false);
  }

  // C/D layout: VGPR r -> lanes 0-15: M=r, lanes 16-31: M=r+8; N=lane%16
  const int orow = m0 + ((lane >> 4) << 3);
  const int ocol = n0 + (lane & 15);
#pragma unroll
  for (int r = 0; r < 8; ++r)
    h[(size_t)(orow + r) * HID_C + ocol] = acc[r];
}

// ---------------- layer-1 aggregation ----------------
__global__ void k_agg_init(const float* __restrict__ h,
                           const float* __restrict__ dinv,
                           float* agg, int n) {
  int idx = blockIdx.x * blockDim.x + threadIdx.x;  // n*HID_C threads
  if (idx >= n * HID_C) return;
  int i = idx >> 6;
  float w = dinv[i];
  agg[idx] = h[idx] * (w * w);  // self-loop contribution
}

// 16 threads per edge, 4 channels each (float4 gather, 4 scalar f32 atomics)
__global__ void k_agg_edges(const long long* __restrict__ ei,
                            const float* __restrict__ h,
                            const float* __restrict__ dinv,
                            float* agg, int E) {
  long long t = (long long)blockIdx.x * blockDim.x + threadIdx.x;
  int e  = (int)(t >> 4);
  if (e >= E) return;
  int c4 = ((int)t & 15) << 2;
  int s = (int)ei[e];
  int d = (int)ei[(size_t)E + e];
  float w = dinv[s] * dinv[d];
  float4 hv = *(const float4*)(h + (size_t)s * HID_C + c4);
  float* dp = agg + (size_t)d * HID_C + c4;
  atomicAdd(dp + 0, hv.x * w);
  atomicAdd(dp + 1, hv.y * w);
  atomicAdd(dp + 2, hv.z * w);
  atomicAdd(dp + 3, hv.w * w);
}

// ---------------- t[i] = relu(agg[i]+b1) . W2 ----------------
__global__ void k_relu_dot(const float* __restrict__ agg,
                           const float* __restrict__ b1,
                           const float* __restrict__ W2,
                           float* tout, int n) {
  int i = blockIdx.x * blockDim.x + threadIdx.x;
  if (i >= n) return;
  const float* a = agg + (size_t)i * HID_C;
  float s = 0.f;
#pragma unroll
  for (int c = 0; c < HID_C; ++c) {
    float v = a[c] + b1[c];
    v = v > 0.f ? v : 0.f;
    s = fmaf(v, W2[c], s);
  }
  tout[i] = s;
}

// ---------------- layer-2 aggregation (scalar channel) ----------------
__global__ void k_out_init(const float* __restrict__ tvec,
                           const float* __restrict__ dinv,
                           const float* __restrict__ b2,
                           float* out, int n) {
  int i = blockIdx.x * blockDim.x + threadIdx.x;
  if (i >= n) return;
  float w = dinv[i];
  out[i] = b2[0] + tvec[i] * (w * w);
}

__global__ void k_out_edges(const long long* __restrict__ ei,
                            const float* __restrict__ tvec,
                            const float* __restrict__ dinv,
                            float* out, int E) {
  int e = blockIdx.x * blockDim.x + threadIdx.x;
  if (e >= E) return;
  int s = (int)ei[e];
  int d = (int)ei[(size_t)E + e];
  atomicAdd(&out[d], tvec[s] * dinv[s] * dinv[d]);
}

extern "C" void kernel_launch(void* const* d_in, const int* in_sizes, int n_in,
                              void* d_out, int out_size, void* d_ws, size_t ws_size,
                              hipStream_t stream) {
  const float*     x  = (const float*)d_in[0];
  const long long* ei = (const long long*)d_in[1];  // int64 [2, E]
  const float*     W1 = (const float*)d_in[2];      // [128, 64]
  const float*     b1 = (const float*)d_in[3];      // [64]
  const float*     W2 = (const float*)d_in[4];      // [64, 1]
  const float*     b2 = (const float*)d_in[5];      // [1]
  float* out = (float*)d_out;

  const int n = in_sizes[0] / IN_C;   // 100000
  const int E = in_sizes[1] / 2;      // 1600000

  // workspace: deg/dinv [n] | h [n*64] | agg [n*64] | t [n]
  float* dinv = (float*)d_ws;
  float* h    = dinv + n;
  float* agg  = h + (size_t)n * HID_C;
  float* tvec = agg + (size_t)n * HID_C;

  const int B = 256;
  // degrees -> dinv (in place)
  k_deg_init<<<(n + B - 1) / B, B, 0, stream>>>(dinv, n);
  k_deg_edges<<<(E + B - 1) / B, B, 0, stream>>>(ei, dinv, E);
  k_dinv<<<(n + B - 1) / B, B, 0, stream>>>(dinv, n);

  // h = x @ W1 on the WMMA pipe (32 rows per block)
  k_gemm1_wmma<<<n / 32, B, 0, stream>>>(x, W1, h, n);

  // layer-1 aggregation
  k_agg_init<<<((size_t)n * HID_C + B - 1) / B, B, 0, stream>>>(h, dinv, agg, n);
  {
    long long tot = (long long)E * 16;
    k_agg_edges<<<(unsigned)((tot + B - 1) / B), B, 0, stream>>>(ei, h, dinv, agg, E);
  }

  // fused relu+bias+dot(W2)
  k_relu_dot<<<(n + B - 1) / B, B, 0, stream>>>(agg, b1, W2, tvec, n);

  // layer-2 aggregation -> out
  k_out_init<<<(n + B - 1) / B, B, 0, stream>>>(tvec, dinv, b2, out, n);
  k_out_edges<<<(E + B - 1) / B, B, 0, stream>>>(ei, tvec, dinv, out, E);
}